// EdgeUpdateNetwork_44555990729422
// MI455X (gfx1250) — compile-verified
//
#include <hip/hip_runtime.h>
#include <math.h>

// Problem constants (reference: T=8, N=128, F=128)
#define TT 8
#define NN 128
#define FF 128

typedef float v4f __attribute__((ext_vector_type(4)));
typedef int   vi4 __attribute__((vector_size(16)));          // matches builtin param
typedef __attribute__((address_space(3))) vi4 lds_vi4;       // LDS-qualified int4

// ---- CDNA5 async global->LDS copy (16B per lane, one wave = one 512B row) ----
// cpol: TH bits [2:0]; 1 = NT (edge stream is read-once, keep L2 for node reuse)
__device__ __forceinline__ void async_copy_b128(const float* g, float* l) {
#if __has_builtin(__builtin_amdgcn_global_load_async_to_lds_b128)
    __builtin_amdgcn_global_load_async_to_lds_b128(
        (vi4*)(void*)g, (lds_vi4*)l, /*offset=*/0, /*cpol=*/1);
#else
    unsigned lds_addr = (unsigned)(unsigned long long)(lds_vi4*)l;
    asm volatile("global_load_async_to_lds_b128 %0, %1, off th:TH_LOAD_NT"
                 :: "v"(lds_addr), "v"((unsigned long long)g)
                 : "memory");
#endif
}

template <int CNT>
__device__ __forceinline__ void wait_asynccnt() {
#if __has_builtin(__builtin_amdgcn_s_wait_asynccnt)
    __builtin_amdgcn_s_wait_asynccnt(CNT);
#else
    asm volatile("s_wait_asynccnt %0" :: "i"(CNT) : "memory");
#endif
}

// ---- CDNA5 hardware tanh (V_TANH_F32 is a native TRANS op on gfx1250) ----
__device__ __forceinline__ float hw_tanhf(float x) {
#if __has_builtin(__builtin_amdgcn_tanhf)
    return __builtin_amdgcn_tanhf(x);
#else
    float y;
    // TRANS hazard rule (ISA 7.4): 1 independent op or V_NOP before the
    // result may be consumed — provide the V_NOP inside the asm.
    asm("v_tanh_f32 %0, %1\n\tv_nop" : "=v"(y) : "v"(x));
    return y;
#endif
}

// One wave processes one output row out[t, j, i, 0:128] per iteration.
// Edge source row edge[t, i, j, 0:128] is streamed through LDS via async
// loads, double-buffered per wave (no cross-wave sharing -> no barriers).
__global__ __launch_bounds__(256, 4)
void edge_update_kernel(const float* __restrict__ node,
                        const float* __restrict__ edge,
                        const float* __restrict__ w,
                        const float* __restrict__ b,
                        float* __restrict__ out,
                        int rows, int wave_stride) {
    __shared__ float buf[8][2][FF];   // 8 waves x 2 buffers x 512B = 8 KB

    const int wave = threadIdx.x >> 5;
    const int lane = threadIdx.x & 31;
    const int f0   = lane * 4;

    const float w0 = w[0], w1 = w[1], w2 = w[2], bb = b[0];
    const float wa = w0 + w1;   // w0*xi + w1*(xi-xj) = (w0+w1)*xi - w1*xj

    float* b0 = &buf[wave][0][0];
    float* b1 = &buf[wave][1][0];

    int r = blockIdx.x * 8 + wave;

    // Prologue: prefetch first row into buffer 0.
    if (r < rows) {
        int t = r >> 14, rem = r & 16383;
        int j = rem >> 7, i = rem & 127;
        long er = ((long)(t * NN + i) * NN + j);
        async_copy_b128(edge + er * FF + f0, b0 + f0);
    }

    int parity = 0;
    for (; r < rows; r += wave_stride) {
        float* cur = parity ? b1 : b0;
        float* nxt = parity ? b0 : b1;

        const int rn = r + wave_stride;
        if (rn < rows) {
            // Prefetch next row, then wait for the *current* one (async loads
            // complete in order => ASYNCcnt<=1 means the older load landed).
            int tn = rn >> 14, remn = rn & 16383;
            int jn = remn >> 7, in = remn & 127;
            long ern = ((long)(tn * NN + in) * NN + jn);
            async_copy_b128(edge + ern * FF + f0, nxt + f0);
            wait_asynccnt<1>();
        } else {
            wait_asynccnt<0>();
        }

        // Decode current row: out row index r = (t*N + j)*N + i
        int t = r >> 14, rem = r & 16383;
        int j = rem >> 7, i = rem & 127;

        v4f e4 = *(const v4f*)(cur + f0);
        v4f ni = *(const v4f*)(node + ((long)(t * NN + i) * FF) + f0);
        v4f nj = *(const v4f*)(node + ((long)(t * NN + j) * FF) + f0);

        v4f o;
        o.x = hw_tanhf(wa * ni.x - w1 * nj.x + w2 * e4.x + bb);
        o.y = hw_tanhf(wa * ni.y - w1 * nj.y + w2 * e4.y + bb);
        o.z = hw_tanhf(wa * ni.z - w1 * nj.z + w2 * e4.z + bb);
        o.w = hw_tanhf(wa * ni.w - w1 * nj.w + w2 * e4.w + bb);
        if (i == j) o = (v4f){0.f, 0.f, 0.f, 0.f};

        // Write-once output: non-temporal so it doesn't evict the edge stream.
        __builtin_nontemporal_store(o, (v4f*)(out + (long)r * FF + f0));

        parity ^= 1;
    }
}

extern "C" void kernel_launch(void* const* d_in, const int* in_sizes, int n_in,
                              void* d_out, int out_size, void* d_ws, size_t ws_size,
                              hipStream_t stream) {
    (void)in_sizes; (void)n_in; (void)out_size; (void)d_ws; (void)ws_size;
    const float* node = (const float*)d_in[0];   // (T,N,F)
    const float* edge = (const float*)d_in[1];   // (T,N,N,F)
    const float* w    = (const float*)d_in[2];   // (3,)
    const float* b    = (const float*)d_in[3];   // (1,)
    float* out = (float*)d_out;                  // (T,N,N,F), transposed in (i,j)

    const int rows   = TT * NN * NN;             // 131072 rows of 128 floats
    const int blocks = 2048;                     // 16384 waves -> 8 rows/wave
    const int wstride = blocks * 8;

    hipLaunchKernelGGL(edge_update_kernel, dim3(blocks), dim3(256), 0, stream,
                       node, edge, w, b, out, rows, wstride);
}